// PSTTransformer_695784702576
// MI455X (gfx1250) — compile-verified
//
#include <hip/hip_runtime.h>

typedef __attribute__((ext_vector_type(2))) float v2f;
typedef __attribute__((ext_vector_type(4))) float v4f;
typedef __attribute__((ext_vector_type(8))) float v8f;

#define WMMA_F32_4(a, b, c) \
  __builtin_amdgcn_wmma_f32_16x16x4_f32(false, (a), false, (b), (short)0, (c), false, false)

#define PI_F        3.14159265358979323846f
#define SCALE_ATT   5.65685424949238f     /* sqrt(32): reference multiplies by dim**0.5 */
#define BN_RSQ      0.9999950000374997f   /* 1/sqrt(1+1e-5) */

// ---------------------------------------------------------------------------
// K1: qkv[b,o,l] = sum_c enc_w[o,c]*xt[b,c,l] + enc_b[o]
//     GEMM M=768, K=256, N=64 per batch; one wave per 16x16 tile; f32 WMMA.
// ---------------------------------------------------------------------------
__global__ void k_qkv(const float* __restrict__ xt, const float* __restrict__ enc_w,
                      const float* __restrict__ enc_b, float* __restrict__ qkv) {
  const int wave = threadIdx.x >> 5;
  const int lane = threadIdx.x & 31;
  const int lm = lane & 15, hi = lane >> 4, koff = hi * 2;
  const int tile = blockIdx.x * 8 + wave;          // 0..3071
  const int b  = tile / 192;
  const int r  = tile % 192;
  const int ot = r >> 2;                           // o-tile (48)
  const int lt = r & 3;                            // l-tile (4)
  const int m = ot * 16 + lm;                      // A row (output channel)
  const int n = lt * 16 + lm;                      // B col (l position)

  const float* wrow = enc_w + (size_t)m * 256;
  const float* xcol = xt + (size_t)b * 256 * 64 + n;

  v8f acc = {};
  for (int k = 0; k < 256; k += 4) {
    v2f a = *(const v2f*)(wrow + k + koff);        // A[m][k+koff], A[m][k+koff+1]
    v2f bb;
    bb.x = xcol[(k + koff) * 64];                  // B[k+koff][n]
    bb.y = xcol[(k + koff + 1) * 64];
    acc = WMMA_F32_4(a, bb, acc);
  }
#pragma unroll
  for (int rr = 0; rr < 8; ++rr) {
    const int o = ot * 16 + rr + 8 * hi;           // C: M = rr(+8), N = lm
    qkv[((size_t)b * 768 + o) * 64 + (lt * 16 + lm)] = acc[rr] + enc_b[o];
  }
}

// ---------------------------------------------------------------------------
// K2: per (b,h): attn = softmax(q k^T * sqrt(32)) v.  q,k,v are (64,32).
//     One 256-thread block (8 waves) per head; WMMA via LDS.
// ---------------------------------------------------------------------------
__global__ void k_attn(const float* __restrict__ qkv, float* __restrict__ attn) {
  __shared__ float qs[64][33];
  __shared__ float ks[64][33];
  __shared__ float vs[64][33];
  __shared__ float ss[64][65];

  const int b = blockIdx.x >> 3, h = blockIdx.x & 7;
  const int tid = threadIdx.x;
  const int wave = tid >> 5, lane = tid & 31;
  const int lm = lane & 15, hi = lane >> 4, koff = hi * 2;
  const size_t baseq = ((size_t)b * 768 + h * 32) * 64;

  // load q/k/v (c-major in memory) transposed into LDS [l][c]
  for (int i = 0; i < 8; ++i) {
    int idx = tid + i * 256;                       // 0..2047
    int c = idx >> 6, l = idx & 63;
    qs[l][c] = qkv[baseq + (size_t)(c)       * 64 + l];
    ks[l][c] = qkv[baseq + (size_t)(256 + c) * 64 + l];
    vs[l][c] = qkv[baseq + (size_t)(512 + c) * 64 + l];
  }
  __syncthreads();

  // scores: (64x64) = q(64x32) * k^T(32x64); 16 tiles, 2 per wave
  for (int it = 0; it < 2; ++it) {
    const int tile = wave * 2 + it;
    const int mt = tile >> 2, nt = tile & 3;
    const int m = mt * 16 + lm, n = nt * 16 + lm;
    v8f acc = {};
    for (int kk = 0; kk < 32; kk += 4) {
      v2f a;  a.x  = qs[m][kk + koff]; a.y  = qs[m][kk + koff + 1];
      v2f bb; bb.x = ks[n][kk + koff]; bb.y = ks[n][kk + koff + 1];
      acc = WMMA_F32_4(a, bb, acc);
    }
#pragma unroll
    for (int rr = 0; rr < 8; ++rr)
      ss[mt * 16 + rr + 8 * hi][nt * 16 + lm] = acc[rr] * SCALE_ATT;
  }
  __syncthreads();

  // softmax over rows of ss
  if (tid < 64) {
    float mx = -3.4e38f;
    for (int j = 0; j < 64; ++j) mx = fmaxf(mx, ss[tid][j]);
    float sm = 0.f;
    for (int j = 0; j < 64; ++j) { float e = expf(ss[tid][j] - mx); ss[tid][j] = e; sm += e; }
    const float inv = 1.f / sm;
    for (int j = 0; j < 64; ++j) ss[tid][j] *= inv;
  }
  __syncthreads();

  // out(64x32) = a(64x64) * v(64x32); 8 tiles, 1 per wave
  {
    const int mt = wave >> 1, nt = wave & 1;
    const int m = mt * 16 + lm, n = nt * 16 + lm;
    v8f acc = {};
    for (int kk = 0; kk < 64; kk += 4) {
      v2f a;  a.x  = ss[m][kk + koff];     a.y  = ss[m][kk + koff + 1];
      v2f bb; bb.x = vs[kk + koff][n];     bb.y = vs[kk + koff + 1][n];
      acc = WMMA_F32_4(a, bb, acc);
    }
#pragma unroll
    for (int rr = 0; rr < 8; ++rr) {
      const int l = mt * 16 + rr + 8 * hi;
      const int c = nt * 16 + lm;
      attn[((size_t)b * 256 + h * 32 + c) * 64 + l] = acc[rr];   // merged heads
    }
  }
}

// ---------------------------------------------------------------------------
// K3: enc_out = sum_m w[c,m]*sin(pi*m/12*attn) + w[c,12+m]*cos(...) + xt
// ---------------------------------------------------------------------------
__global__ void k_fourier(const float* __restrict__ attn, const float* __restrict__ wts,
                          const float* __restrict__ xt, float* __restrict__ enc_out) {
  const int t = blockIdx.x * 256 + threadIdx.x;    // 262144 total, exact
  const int c = (t >> 6) & 255;
  const float a = attn[t];
  float acc = 0.f;
#pragma unroll
  for (int m = 0; m < 12; ++m) {
    const float ph = ((float)m * (PI_F / 12.0f)) * a;
    float s, co;
    __sincosf(ph, &s, &co);
    acc += wts[c * 24 + m] * s + wts[c * 24 + 12 + m] * co;
  }
  enc_out[t] = acc + xt[t];
}

// ---------------------------------------------------------------------------
// K4: decoder, reduced. Query vector is constant over N and B (mask_token
//     broadcast), so the decoder output is one 256-vector per batch.
// ---------------------------------------------------------------------------
__global__ void k_decoder(const float* __restrict__ enc_out, const float* __restrict__ dec_w,
                          const float* __restrict__ dec_b, const float* __restrict__ mask,
                          float* __restrict__ dec_vec) {
  __shared__ float qh[32];
  __shared__ float sc[64];
  const int b = blockIdx.x >> 3, h = blockIdx.x & 7;
  const int tid = threadIdx.x;                     // 64 threads

  if (tid < 32) {                                  // q = dec_w @ mask + dec_b
    float acc = dec_b[h * 32 + tid];
    const float* wr = dec_w + (size_t)(h * 32 + tid) * 256;
    for (int c = 0; c < 256; ++c) acc += wr[c] * mask[c];
    qh[tid] = acc;
  }
  __syncthreads();
  {                                                // scores over 64 latents
    float s = 0.f;
    const float* kv = enc_out + ((size_t)b * 256 + h * 32) * 64 + tid;
    for (int c = 0; c < 32; ++c) s += qh[c] * kv[c * 64];
    sc[tid] = s * SCALE_ATT;
  }
  __syncthreads();
  if (tid == 0) {                                  // softmax (tiny, serial)
    float mx = -3.4e38f;
    for (int j = 0; j < 64; ++j) mx = fmaxf(mx, sc[j]);
    float sm = 0.f;
    for (int j = 0; j < 64; ++j) { float e = expf(sc[j] - mx); sc[j] = e; sm += e; }
    const float inv = 1.f / sm;
    for (int j = 0; j < 64; ++j) sc[j] *= inv;
  }
  __syncthreads();
  if (tid < 32) {                                  // out = a @ kv + mask residual
    const float* kv = enc_out + ((size_t)b * 256 + h * 32 + tid) * 64;
    float o = 0.f;
    for (int s = 0; s < 64; ++s) o += sc[s] * kv[s];
    dec_vec[b * 256 + h * 32 + tid] = o + mask[h * 32 + tid];
  }
}

// ---------------------------------------------------------------------------
// K5: y_vec[b,0..2] = lrelu(bn(fc2 @ lrelu(bn(fc1 @ dec_vec))))
// ---------------------------------------------------------------------------
__global__ void k_head(const float* __restrict__ dec_vec,
                       const float* __restrict__ fc1_w, const float* __restrict__ fc1_g,
                       const float* __restrict__ fc1_b,
                       const float* __restrict__ fc2_w, const float* __restrict__ fc2_g,
                       const float* __restrict__ fc2_b,
                       float* __restrict__ y_vec) {
  __shared__ float xv[256];
  __shared__ float y1[256];
  const int b = blockIdx.x, tid = threadIdx.x;
  xv[tid] = dec_vec[b * 256 + tid];
  __syncthreads();
  float acc = 0.f;
  const float* wr = fc1_w + (size_t)tid * 256;
  for (int c = 0; c < 256; ++c) acc += wr[c] * xv[c];
  float y = fc1_g[tid] * acc * BN_RSQ + fc1_b[tid];
  y1[tid] = (y >= 0.f) ? y : 0.2f * y;
  __syncthreads();
  if (tid < 3) {
    float a2 = 0.f;
    const float* w2 = fc2_w + tid * 256;
    for (int c = 0; c < 256; ++c) a2 += w2[c] * y1[c];
    float z = fc2_g[tid] * a2 * BN_RSQ + fc2_b[tid];
    y_vec[b * 3 + tid] = (z >= 0.f) ? z : 0.2f * z;
  }
}

// ---------------------------------------------------------------------------
// K6: broadcast writes (the 135.8 MB bandwidth-bound part), b128 stores.
//     out = [ xn_pre (B,C,N) const over N | y (B,N,3) const over N ]
// ---------------------------------------------------------------------------
__global__ void k_broadcast(const float* __restrict__ dec_vec, const float* __restrict__ y_vec,
                            float* __restrict__ out) {
  const int blk = blockIdx.x, tid = threadIdx.x;
  if (blk < 4096) {                                // one (b,c) row of 8192 per block
    const float v = dec_vec[blk];
    v4f v4 = {v, v, v, v};
    v4f* dst = (v4f*)out + (size_t)blk * 2048;
#pragma unroll
    for (int i = 0; i < 8; ++i) dst[tid + i * 256] = v4;
  } else {                                         // y region: 98304 float4 over 48 blocks
    const int yb = blk - 4096;
    const size_t base4 = (size_t)yb * 2048;
    v4f* dst = (v4f*)out + 8388608;                // 33,554,432 floats / 4
#pragma unroll
    for (int i = 0; i < 8; ++i) {
      const size_t f = base4 + tid + i * 256;
      const size_t fi = f * 4;
      const int bb  = (int)(fi / 24576);           // 8192*3 per batch (divisible by 4)
      const int rem = (int)(fi % 24576);
      v4f v;
      v.x = y_vec[bb * 3 + (rem    ) % 3];
      v.y = y_vec[bb * 3 + (rem + 1) % 3];
      v.z = y_vec[bb * 3 + (rem + 2) % 3];
      v.w = y_vec[bb * 3 + (rem + 3) % 3];
      dst[f] = v;
    }
  }
}

// ---------------------------------------------------------------------------
extern "C" void kernel_launch(void* const* d_in, const int* in_sizes, int n_in,
                              void* d_out, int out_size, void* d_ws, size_t ws_size,
                              hipStream_t stream) {
  const float* xt      = (const float*)d_in[0];
  /* d_in[1] = xn : unused (fully masked path) */
  const float* weights = (const float*)d_in[2];
  const float* mask    = (const float*)d_in[3];
  const float* enc_w   = (const float*)d_in[4];
  const float* enc_b   = (const float*)d_in[5];
  const float* dec_w   = (const float*)d_in[6];
  const float* dec_b   = (const float*)d_in[7];
  const float* fc1_w   = (const float*)d_in[8];
  const float* fc1_g   = (const float*)d_in[9];
  const float* fc1_b   = (const float*)d_in[10];
  const float* fc2_w   = (const float*)d_in[11];
  const float* fc2_g   = (const float*)d_in[12];
  const float* fc2_b   = (const float*)d_in[13];

  float* out = (float*)d_out;
  float* ws  = (float*)d_ws;
  float* qkv     = ws;                 // 786432 floats
  float* attn    = ws + 786432;        // 262144
  float* enc_out = ws + 1048576;       // 262144
  float* dec_vec = ws + 1310720;       // 4096
  float* y_vec   = ws + 1314816;       // 48

  hipLaunchKernelGGL(k_qkv,       dim3(384),  dim3(256), 0, stream, xt, enc_w, enc_b, qkv);
  hipLaunchKernelGGL(k_attn,      dim3(128),  dim3(256), 0, stream, qkv, attn);
  hipLaunchKernelGGL(k_fourier,   dim3(1024), dim3(256), 0, stream, attn, weights, xt, enc_out);
  hipLaunchKernelGGL(k_decoder,   dim3(128),  dim3(64),  0, stream, enc_out, dec_w, dec_b, mask, dec_vec);
  hipLaunchKernelGGL(k_head,      dim3(16),   dim3(256), 0, stream, dec_vec,
                     fc1_w, fc1_g, fc1_b, fc2_w, fc2_g, fc2_b, y_vec);
  hipLaunchKernelGGL(k_broadcast, dim3(4144), dim3(256), 0, stream, dec_vec, y_vec, out);
}